// ERR_11493332484555
// MI455X (gfx1250) — compile-verified
//
#include <hip/hip_runtime.h>
#include <hip/hip_bf16.h>

// ---- problem constants (from reference) ----
#define BB     2
#define HH     64
#define WW     64
#define LL     4096         // HH*WW
#define CMODEL 96
#define DIN    192          // 2*CMODEL
#define KDIR   4
#define NST    16           // d_state
#define RDT    6            // dt_rank
#define CDBL   38           // RDT + 2*NST
#define EPROJ  384          // 2*DIN
#define STILE  32           // scan LDS tile (steps)

typedef __attribute__((ext_vector_type(2))) float v2f;
typedef __attribute__((ext_vector_type(8))) float v8f;

// V_WMMA_F32_16X16X4_F32 : D = A(16x4, f32) * B(4x16, f32) + C(16x16, f32)
__device__ __forceinline__ v8f wmma4(v2f a, v2f b, v8f c) {
  return __builtin_amdgcn_wmma_f32_16x16x4_f32(
      /*neg_a=*/false, a, /*neg_b=*/false, b,
      /*c_mod=*/(short)0, c, /*reuse_a=*/false, /*reuse_b=*/false);
}

// ---------------- 1) in_proj GEMM: (8192x96) @ (96x384) -> xi(192) | z(192) ----
__global__ __launch_bounds__(256) void k_inproj(const float* __restrict__ x,
                                                const float* __restrict__ w,
                                                float* __restrict__ xi,
                                                float* __restrict__ z) {
  const int NT = EPROJ / 16;                       // 24 col tiles
  int wave = (blockIdx.x * blockDim.x + threadIdx.x) >> 5;
  int lane = threadIdx.x & 31;
  int mt = wave / NT, nt = wave % NT;
  int m = mt * 16 + (lane & 15);
  int n = nt * 16 + (lane & 15);
  int kb = (lane >> 4) << 1;                       // 0 or 2
  v8f acc = {};
  for (int kk = 0; kk < CMODEL; kk += 4) {
    int k0 = kk + kb;
    v2f a, b;
    a.x = x[m * CMODEL + k0];
    a.y = x[m * CMODEL + k0 + 1];
    b.x = w[n * CMODEL + k0];                      // B[k][n] = W[n][k]
    b.y = w[n * CMODEL + k0 + 1];
    acc = wmma4(a, b, acc);
  }
  int row0 = mt * 16 + ((lane >> 4) << 3);
#pragma unroll
  for (int v = 0; v < 8; ++v) {
    int row = row0 + v;
    float val = acc[v];
    if (n < DIN) xi[row * DIN + n] = val;
    else         z[row * DIN + (n - DIN)] = val;
  }
}

// -------- 2) depthwise 3x3 conv + bias + SiLU; writes xc (h-major) + xcT -------
__global__ __launch_bounds__(256) void k_conv(const float* __restrict__ xi,
                                              const float* __restrict__ cw,
                                              const float* __restrict__ cb,
                                              float* __restrict__ xc,
                                              float* __restrict__ xcT) {
  int idx = blockIdx.x * blockDim.x + threadIdx.x;   // B*D*L
  int l = idx & (LL - 1);
  int t = idx >> 12;
  int d = t % DIN;
  int b = t / DIN;
  int h = l >> 6, w = l & 63;
  float s = cb[d];
#pragma unroll
  for (int kh = 0; kh < 3; ++kh) {
#pragma unroll
    for (int kw = 0; kw < 3; ++kw) {
      int hh = h + kh - 1, ww = w + kw - 1;
      if (hh >= 0 && hh < HH && ww >= 0 && ww < WW)
        s += xi[((size_t)b * LL + hh * WW + ww) * DIN + d] * cw[d * 9 + kh * 3 + kw];
    }
  }
  float sil = s / (1.f + __expf(-s));
  size_t plane = ((size_t)b * DIN + d) * LL;
  xc [plane + h * WW + w] = sil;                     // row-major scan order
  xcT[plane + w * HH + h] = sil;                     // col-major scan order
}

// ---------------- 3) x_proj GEMM per (b,k): (38x192) @ (192xL) -----------------
__global__ __launch_bounds__(256) void k_xproj(const float* __restrict__ xpw,
                                               const float* __restrict__ xc,
                                               const float* __restrict__ xcT,
                                               float* __restrict__ xdbl) {
  int wave = (blockIdx.x * blockDim.x + threadIdx.x) >> 5;
  int lane = threadIdx.x & 31;
  int lt = wave & 255;                // 256 L-tiles
  int t2 = wave >> 8;
  int ct = t2 % 3;                    // 3 row tiles (38 padded to 48)
  int bk = t2 / 3;
  int b = bk >> 2, k = bk & 3;
  int c = ct * 16 + (lane & 15);
  int l = lt * 16 + (lane & 15);
  int kb = (lane >> 4) << 1;
  int g = (k >= 2) ? (LL - 1 - l) : l;               // contiguous (maybe reversed)
  const float* W = xpw + (size_t)k * CDBL * DIN;
  const float* X = ((k & 1) ? xcT : xc) + (size_t)b * DIN * LL;
  int cc = (c < CDBL) ? c : 0;        // clamp to keep loads in-bounds
  bool cv = (c < CDBL);
  v8f acc = {};
  for (int kk = 0; kk < DIN; kk += 4) {
    int k0 = kk + kb;
    v2f a, bb;
    float a0 = W[cc * DIN + k0];
    float a1 = W[cc * DIN + k0 + 1];
    a.x = cv ? a0 : 0.f;
    a.y = cv ? a1 : 0.f;
    bb.x = X[(size_t)k0 * LL + g];
    bb.y = X[(size_t)(k0 + 1) * LL + g];
    acc = wmma4(a, bb, acc);
  }
  float* O = xdbl + (size_t)bk * CDBL * LL;
  int row0 = ct * 16 + ((lane >> 4) << 3);
#pragma unroll
  for (int v = 0; v < 8; ++v) {
    int row = row0 + v;
    if (row < CDBL) O[(size_t)row * LL + l] = acc[v];
  }
}

// ---------------- 4) dt projection (r=6) + softplus -> delta (bk,d,l) ----------
__global__ __launch_bounds__(256) void k_delta(const float* __restrict__ xdbl,
                                               const float* __restrict__ dtw,
                                               const float* __restrict__ dtb,
                                               float* __restrict__ delta) {
  int idx = blockIdx.x * blockDim.x + threadIdx.x;   // B*K*D*L
  int l = idx & (LL - 1);
  int t = idx >> 12;
  int d = t % DIN;
  int bk = t / DIN;
  int k = bk & 3;
  const float* Wd = dtw + (size_t)(k * DIN + d) * RDT;
  const float* Xd = xdbl + (size_t)bk * CDBL * LL + l;
  float s = dtb[k * DIN + d];
#pragma unroll
  for (int r = 0; r < RDT; ++r) s += Wd[r] * Xd[(size_t)r * LL];
  float sp = (s > 20.f) ? s : log1pf(__expf(s));
  delta[idx] = sp;
}

// ---- 5) selective scan: thread per (b,k,d); B/C tiles copied to LDS with
//         GLOBAL_LOAD_ASYNC_TO_LDS_B128 (ASYNCcnt), double-buffered ------------
__global__ __launch_bounds__(192) void k_scan(const float* __restrict__ delta,
                                              const float* __restrict__ xc,
                                              const float* __restrict__ xcT,
                                              const float* __restrict__ xdbl,
                                              const float* __restrict__ alogs,
                                              const float* __restrict__ dsv,
                                              float* __restrict__ oy) {
  __shared__ float sBC[2][2 * NST][STILE];           // 8 KB, double-buffered
  int bk = blockIdx.x;                               // B*K = 8 blocks
  int b = bk >> 2, k = bk & 3;
  int d = threadIdx.x;                               // 192 channels

  float A[NST], h[NST];
#pragma unroll
  for (int n = 0; n < NST; ++n) {
    A[n] = -__expf(alogs[(size_t)(k * DIN + d) * NST + n]);
    h[n] = 0.f;
  }
  float Dv = dsv[k * DIN + d];
  const float* dp   = delta + ((size_t)bk * DIN + d) * LL;
  const float* xsrc = ((k & 1) ? xcT : xc) + ((size_t)b * DIN + d) * LL;
  const bool rev = (k >= 2);
  const float* bc = xdbl + (size_t)bk * CDBL * LL + (size_t)RDT * LL; // B rows, then C
  float* op = oy + ((size_t)bk * DIN + d) * LL;

  // waves 0-3 copy one 4KB tile (32 rows x 32 floats) straight into LDS:
  // 256 x 16B transfers = 128 lanes x 2 rounds -> exactly 2 async instr / wave.
  // The LDS destination offset is derived from &sBC[...] (ptrtoint): bits[31:0]
  // of a generic shared pointer are the workgroup-relative LDS byte address
  // (ISA 10.2), and the capture makes LLVM treat the asm as a writer of sBC.
  auto issue_async = [&](int buf, int l0) {
    if (threadIdx.x < 128) {
#pragma unroll
      for (int rnd = 0; rnd < 2; ++rnd) {
        int i = threadIdx.x + rnd * 128;             // 0..255
        int row = i >> 3;                            // 0..31 (16 B rows + 16 C rows)
        int c4  = (i & 7) << 2;                      // float column, 16B aligned
        const float* g = bc + (size_t)row * LL + l0 + c4;
        unsigned lds = (unsigned)(size_t)&sBC[buf][row][c4];
        asm volatile("global_load_async_to_lds_b128 %0, %1, off"
                     :: "v"(lds), "v"(g) : "memory");
      }
    }
  };

  issue_async(0, 0);
  const int NTILES = LL / STILE;                     // 128
  for (int t = 0; t < NTILES; ++t) {
    int buf = t & 1;
    int l0 = t * STILE;
    if (t + 1 < NTILES) {
      issue_async(buf ^ 1, l0 + STILE);              // prefetch next tile
      asm volatile("s_wait_asynccnt 0x2" ::: "memory"); // tile t landed (in-order)
    } else {
      asm volatile("s_wait_asynccnt 0x0" ::: "memory");
    }
    __syncthreads();                                 // publish tile t to all waves

    // per-thread delta tile, vector-loaded (128B aligned)
    float dts[STILE];
    const float4* dp4 = reinterpret_cast<const float4*>(dp + l0);
#pragma unroll
    for (int q = 0; q < STILE / 4; ++q) {
      float4 v = dp4[q];
      dts[4 * q] = v.x; dts[4 * q + 1] = v.y; dts[4 * q + 2] = v.z; dts[4 * q + 3] = v.w;
    }

    for (int j = 0; j < STILE; ++j) {
      int l = l0 + j;
      float dt = dts[j];
      float u  = xsrc[rev ? (LL - 1 - l) : l];
      float du = dt * u;
      float y = 0.f;
#pragma unroll
      for (int n = 0; n < NST; ++n) {
        float dA = __expf(dt * A[n]);                // v_exp_f32 (TRANS)
        h[n] = h[n] * dA + du * sBC[buf][n][j];      // LDS broadcast read
        y += h[n] * sBC[buf][NST + n][j];
      }
      op[l] = y + u * Dv;
    }
    __syncthreads();                                 // all reads of buf done before
  }                                                  // next iter overwrites it
}

// ---------------- 6) merge 4 directions back to row-major (b,l,d) --------------
__global__ __launch_bounds__(256) void k_merge(const float* __restrict__ oy,
                                               float* __restrict__ ym) {
  int idx = blockIdx.x * blockDim.x + threadIdx.x;   // B*L*D as (b,l,d)
  int d = idx % DIN;
  int t = idx / DIN;
  int l = t & (LL - 1);
  int b = t >> 12;
  int p1 = (l & 63) * 64 + (l >> 6);                 // H<->W transposed position
  const float* base = oy + (size_t)b * KDIR * DIN * LL;
  float v = base[(size_t)(0 * DIN + d) * LL + l]
          + base[(size_t)(2 * DIN + d) * LL + (LL - 1 - l)]
          + base[(size_t)(1 * DIN + d) * LL + p1]
          + base[(size_t)(3 * DIN + d) * LL + (LL - 1 - p1)];
  ym[idx] = v;
}

// ---------------- 7a) LayerNorm(192) + SiLU(z) gate (in-place safe) ------------
__global__ __launch_bounds__(256) void k_lngate(float* __restrict__ y,
                                                const float* __restrict__ z,
                                                const float* __restrict__ nw,
                                                const float* __restrict__ nb,
                                                float* __restrict__ yng) {
  int p = blockIdx.x * blockDim.x + threadIdx.x;     // B*L pixels
  const float* yp = y + (size_t)p * DIN;
  const float* zp = z + (size_t)p * DIN;
  float s = 0.f, s2 = 0.f;
  for (int d = 0; d < DIN; ++d) { float v = yp[d]; s += v; s2 += v * v; }
  float mu = s * (1.f / DIN);
  float var = s2 * (1.f / DIN) - mu * mu;
  float inv = rsqrtf(var + 1e-5f);
  for (int d = 0; d < DIN; ++d) {
    float yn = (yp[d] - mu) * inv * nw[d] + nb[d];
    float zv = zp[d];
    float g = zv / (1.f + __expf(-zv));
    yng[(size_t)p * DIN + d] = yn * g;
  }
}

// ---------------- 7b) out_proj GEMM: (8192x192) @ (192x96) -> d_out ------------
__global__ __launch_bounds__(256) void k_outproj(const float* __restrict__ yng,
                                                 const float* __restrict__ w,
                                                 float* __restrict__ out) {
  const int NT = CMODEL / 16;                        // 6 col tiles
  int wave = (blockIdx.x * blockDim.x + threadIdx.x) >> 5;
  int lane = threadIdx.x & 31;
  int mt = wave / NT, nt = wave % NT;
  int m = mt * 16 + (lane & 15);
  int n = nt * 16 + (lane & 15);
  int kb = (lane >> 4) << 1;
  v8f acc = {};
  for (int kk = 0; kk < DIN; kk += 4) {
    int k0 = kk + kb;
    v2f a, b;
    a.x = yng[(size_t)m * DIN + k0];
    a.y = yng[(size_t)m * DIN + k0 + 1];
    b.x = w[n * DIN + k0];                           // B[k][n] = W[n][k]
    b.y = w[n * DIN + k0 + 1];
    acc = wmma4(a, b, acc);
  }
  int row0 = mt * 16 + ((lane >> 4) << 3);
#pragma unroll
  for (int v = 0; v < 8; ++v)
    out[(size_t)(row0 + v) * CMODEL + n] = acc[v];
}

extern "C" void kernel_launch(void* const* d_in, const int* in_sizes, int n_in,
                              void* d_out, int out_size, void* d_ws, size_t ws_size,
                              hipStream_t stream) {
  const float* x     = (const float*)d_in[0];
  const float* inw   = (const float*)d_in[1];
  const float* convw = (const float*)d_in[2];
  const float* convb = (const float*)d_in[3];
  const float* xpw   = (const float*)d_in[4];
  const float* dtw   = (const float*)d_in[5];
  const float* dtb   = (const float*)d_in[6];
  const float* alogs = (const float*)d_in[7];
  const float* dsv   = (const float*)d_in[8];
  const float* nw    = (const float*)d_in[9];
  const float* nb    = (const float*)d_in[10];
  const float* opw   = (const float*)d_in[11];
  float* out = (float*)d_out;
  float* ws  = (float*)d_ws;

  constexpr size_t SZ_TOK = (size_t)BB * LL * DIN;           // 1,572,864 floats
  constexpr size_t SZ_BIG = (size_t)BB * KDIR * DIN * LL;    // 6,291,456 floats
  // region0: xi (then reused for x_dbl, which is smaller: B*K*38*L)
  float* xi    = ws;
  float* xdbl  = ws;                                         // alias (xi dead)
  float* zg    = ws + SZ_TOK;
  float* xc    = ws + 2 * SZ_TOK;
  float* xcT   = ws + 3 * SZ_TOK;
  float* delta = ws + 4 * SZ_TOK;                            // SZ_BIG
  float* oy    = ws + 4 * SZ_TOK + SZ_BIG;                   // SZ_BIG
  float* ym    = delta;                                      // alias (delta dead)
  float* yng   = ym;                                         // in-place (safe)

  // 1) in_proj: 512*24 wave tiles, 8 waves/block
  k_inproj<<<(512 * 24) / 8, 256, 0, stream>>>(x, inw, xi, zg);
  // 2) conv+SiLU (dual-layout output): B*D*L threads
  k_conv<<<(BB * DIN * LL) / 256, 256, 0, stream>>>(xi, convw, convb, xc, xcT);
  // 3) x_proj: B*K*3*256 wave tiles
  k_xproj<<<(BB * KDIR * 3 * 256) / 8, 256, 0, stream>>>(xpw, xc, xcT, xdbl);
  // 4) delta: B*K*D*L threads
  k_delta<<<(BB * KDIR * DIN * LL) / 256, 256, 0, stream>>>(xdbl, dtw, dtb, delta);
  // 5) scan: one block per (b,k), one thread per channel
  k_scan<<<BB * KDIR, DIN, 0, stream>>>(delta, xc, xcT, xdbl, alogs, dsv, oy);
  // 6) merge directions
  k_merge<<<(BB * LL * DIN) / 256, 256, 0, stream>>>(oy, ym);
  // 7) layernorm + gate, then out_proj
  k_lngate<<<(BB * LL) / 256, 256, 0, stream>>>(ym, zg, nw, nb, yng);
  k_outproj<<<(512 * 6) / 8, 256, 0, stream>>>(yng, opw, out);
  (void)in_sizes; (void)n_in; (void)out_size; (void)ws_size;
}